// StructuralProbe_9405978378446
// MI455X (gfx1250) — compile-verified
//
#include <hip/hip_runtime.h>

// Problem constants (match reference)
#define Bn 64
#define Ln 512
#define En 1024
#define Dn 128

typedef __attribute__((ext_vector_type(2))) float v2f;
typedef __attribute__((ext_vector_type(8))) float v8f;

// ---------------------------------------------------------------------------
// Kernel 1: dep = emb @ W^T   (M=32768, N=128, K=1024, fp32 WMMA 16x16x4)
// Block = 256 threads = 8 waves. Wave w computes N-tile [w*16, w*16+16);
// the 8 waves cover all of N=128. Block owns an M-tile of 64 rows: each wave
// keeps FOUR 16x16 accumulators that share a single B fragment per k-step
// (5 global_load_b64 per 4 v_wmma instead of 3 per 2), and the four
// independent accumulation chains give the scheduler latency-hiding room.
//
// fp32 WMMA fragment layout (ISA 7.12.2):
//   A 16x4:  lanes 0-15 hold M=lane rows; VGPR0 = K=k0+2*kh, VGPR1 = K+1
//            where kh = lane>=16 -> one float2 load at emb[row*K + k0+2*kh].
//   B 4x16:  mirrored on N: B[k][n] = W[n][k] -> same float2 pattern on W.
//   C/D 16x16: VGPR r -> row r + 8*kh, col = lane%16.
// ---------------------------------------------------------------------------
__global__ __launch_bounds__(256) void dep_gemm_kernel(
    const float* __restrict__ emb, const float* __restrict__ W,
    float* __restrict__ dep) {
  const int lane  = threadIdx.x & 31;
  const int wave  = threadIdx.x >> 5;
  const int r16   = lane & 15;
  const int kh    = lane >> 4;           // 0 or 1 (K half)
  const int n_base = wave * 16;          // 8 waves cover N = 128
  const int m_base = blockIdx.x * 64;    // four 16-row tiles per block

  const float* arow0 = emb + (size_t)(m_base + r16) * En;
  const float* arow1 = arow0 + (size_t)16 * En;
  const float* arow2 = arow0 + (size_t)32 * En;
  const float* arow3 = arow0 + (size_t)48 * En;
  const float* brow  = W   + (size_t)(n_base + r16) * En;

  v8f c0 = {}, c1 = {}, c2 = {}, c3 = {};

#pragma unroll 4
  for (int k0 = 0; k0 < En; k0 += 4) {
    const int kk = k0 + kh * 2;
    v2f bf = *(const v2f*)(brow  + kk);   // shared B fragment
    v2f a0 = *(const v2f*)(arow0 + kk);
    v2f a1 = *(const v2f*)(arow1 + kk);
    v2f a2 = *(const v2f*)(arow2 + kk);
    v2f a3 = *(const v2f*)(arow3 + kk);
    // D = A*B + C ; 8 args: (neg_a, A, neg_b, B, c_mod, C, reuse_a, reuse_b)
    c0 = __builtin_amdgcn_wmma_f32_16x16x4_f32(false, a0, false, bf,
                                               (short)0, c0, false, false);
    c1 = __builtin_amdgcn_wmma_f32_16x16x4_f32(false, a1, false, bf,
                                               (short)0, c1, false, false);
    c2 = __builtin_amdgcn_wmma_f32_16x16x4_f32(false, a2, false, bf,
                                               (short)0, c2, false, false);
    c3 = __builtin_amdgcn_wmma_f32_16x16x4_f32(false, a3, false, bf,
                                               (short)0, c3, false, false);
  }

  // Store per C/D layout: row = m_base + t*16 + kh*8 + r, col = n_base + r16.
  float* out0 = dep + (size_t)(m_base + kh * 8) * Dn + n_base + r16;
#pragma unroll
  for (int r = 0; r < 8; ++r) {
    out0[(size_t)r * Dn]        = c0[r];
    out0[(size_t)(16 + r) * Dn] = c1[r];
    out0[(size_t)(32 + r) * Dn] = c2[r];
    out0[(size_t)(48 + r) * Dn] = c3[r];
  }
}

// ---------------------------------------------------------------------------
// Kernel 2: sq[b,i] = sum_o dep[b,i,o]^2. One wave per row (D=128 = 32 float4).
// ---------------------------------------------------------------------------
__global__ __launch_bounds__(256) void rownorm_kernel(
    const float* __restrict__ dep, float* __restrict__ sq) {
  const int row  = blockIdx.x * 8 + (threadIdx.x >> 5);
  const int lane = threadIdx.x & 31;
  const float4* p = (const float4*)(dep + (size_t)row * Dn);
  float4 v = p[lane];
  float s = v.x * v.x + v.y * v.y + v.z * v.z + v.w * v.w;
#pragma unroll
  for (int off = 16; off > 0; off >>= 1)
    s += __shfl_xor(s, off, 32);
  if (lane == 0) sq[row] = s;
}

// ---------------------------------------------------------------------------
// Kernel 3: distances[b,i,j] = max(sq_i + sq_j - 2 * dep_i . dep_j, 0)
// Each wave computes a 16x32 (i,j) tile: two 16x16 accumulators share one A
// fragment per k-step (3 loads per 2 wmma), and the two chains are
// independent. Block = 8 waves sharing the same i-tile (A rows hot in WGP$),
// covering a 256-wide j-stripe. K=128 -> 32 k-steps per accumulator.
// ---------------------------------------------------------------------------
__global__ __launch_bounds__(256) void dist_kernel(
    const float* __restrict__ dep, const float* __restrict__ sq,
    float* __restrict__ dist) {
  const int lane  = threadIdx.x & 31;
  const int wave  = threadIdx.x >> 5;
  const int r16   = lane & 15;
  const int kh    = lane >> 4;
  const int b     = blockIdx.z;
  const int i_base = blockIdx.x * 16;
  const int j_base = blockIdx.y * 256 + wave * 32;

  const float* depb = dep + (size_t)b * Ln * Dn;
  const float* sqb  = sq  + (size_t)b * Ln;
  const float* arow  = depb + (size_t)(i_base + r16) * Dn;
  const float* brow0 = depb + (size_t)(j_base + r16) * Dn;
  const float* brow1 = brow0 + (size_t)16 * Dn;

  v8f ca = {}, cb = {};
#pragma unroll
  for (int k0 = 0; k0 < Dn; k0 += 4) {
    const int kk = k0 + kh * 2;
    v2f a  = *(const v2f*)(arow  + kk);   // shared A fragment
    v2f b0 = *(const v2f*)(brow0 + kk);
    v2f b1 = *(const v2f*)(brow1 + kk);
    ca = __builtin_amdgcn_wmma_f32_16x16x4_f32(false, a, false, b0,
                                               (short)0, ca, false, false);
    cb = __builtin_amdgcn_wmma_f32_16x16x4_f32(false, a, false, b1,
                                               (short)0, cb, false, false);
  }

  const float sqj0 = sqb[j_base + r16];        // column norms (per-lane)
  const float sqj1 = sqb[j_base + 16 + r16];
  float* out = dist + (size_t)b * Ln * Ln +
               (size_t)(i_base + kh * 8) * Ln + j_base + r16;
#pragma unroll
  for (int r = 0; r < 8; ++r) {
    const float sqi = sqb[i_base + kh * 8 + r];
    out[(size_t)r * Ln]      = fmaxf(sqi + sqj0 - 2.0f * ca[r], 0.0f);
    out[(size_t)r * Ln + 16] = fmaxf(sqi + sqj1 - 2.0f * cb[r], 0.0f);
  }
}

// ---------------------------------------------------------------------------
extern "C" void kernel_launch(void* const* d_in, const int* in_sizes, int n_in,
                              void* d_out, int out_size, void* d_ws,
                              size_t ws_size, hipStream_t stream) {
  const float* emb = (const float*)d_in[0];  // (B, L, E) fp32
  const float* W   = (const float*)d_in[1];  // (D, E)    fp32

  float* dep  = (float*)d_out;                       // (B, L, D)
  float* dist = dep + (size_t)Bn * Ln * Dn;          // (B, L, L)
  float* sq   = (float*)d_ws;                        // (B, L) = 128 KB scratch

  // 1) dep = emb @ W^T : 512 blocks x 256 threads (M-tile 64, full N per block)
  dep_gemm_kernel<<<dim3((Bn * Ln) / 64), 256, 0, stream>>>(emb, W, dep);

  // 2) row squared norms
  rownorm_kernel<<<dim3((Bn * Ln) / 8), 256, 0, stream>>>(dep, sq);

  // 3) pairwise squared distances (j-stripe of 256 per block)
  dist_kernel<<<dim3(Ln / 16, Ln / 256, Bn), 256, 0, stream>>>(dep, sq, dist);
}